// GridSample_83021717831974
// MI455X (gfx1250) — compile-verified
//
#include <hip/hip_runtime.h>

// Bilinear grid-sample (zeros padding), matching the JAX reference:
//   x   : [N=8,  C=64, H=256, W=256] f32
//   grid: [N=8, Ho=256, Wo=256, 2]   f32  (dx, dy pixel offsets)
//   out : [N,C,Ho,Wo] f32
//
// Strategy (MI455X):
//  - one block per (n, ho) row, lane = wo  -> coalesced NT stores, near-linear gathers
//  - per-pixel coords/weights computed once, reused across all 64 channels
//  - output stored non-temporally so the 134MB input stays resident in the 192MB L2
//  - light prefetch (once per 8-channel group) to warm caches without eating
//    VMEM issue slots (2 SIMDs per CU share one memory path)

namespace {
constexpr int Nn = 8;
constexpr int Cc = 64;
constexpr int Hh = 256;
constexpr int Ww = 256;
constexpr int HW = Hh * Ww;
}

typedef float v2f __attribute__((ext_vector_type(2)));

__global__ __launch_bounds__(256) void grid_sample_bilinear_zeros(
    const float* __restrict__ x, const float* __restrict__ grid,
    float* __restrict__ out)
{
    const int wo = threadIdx.x;               // 0..255 (column)
    const int ho = blockIdx.x & (Hh - 1);     // row
    const int n  = blockIdx.x >> 8;           // batch

    // grid[n, ho, wo, 0:2] -- read once, non-temporal (never reused)
    const v2f* gptr = reinterpret_cast<const v2f*>(grid)
                    + ((size_t)(n * Hh + ho) * Ww + wo);
    const v2f g = __builtin_nontemporal_load(gptr);

    // Replicate the reference FP chain exactly (normalize -> unnormalize).
    const float abs_x = g.x + (float)wo;
    const float abs_y = g.y + (float)ho;
    const float norm_x = abs_x * 2.0f / (float)(Ww - 1) - 1.0f;
    const float norm_y = abs_y * 2.0f / (float)(Hh - 1) - 1.0f;
    const float ix = (norm_x + 1.0f) * 0.5f * (float)(Ww - 1);
    const float iy = (norm_y + 1.0f) * 0.5f * (float)(Hh - 1);

    const float x0f = floorf(ix);
    const float y0f = floorf(iy);
    const float x1f = x0f + 1.0f;
    const float y1f = y0f + 1.0f;

    const float wx1 = ix - x0f, wx0 = 1.0f - wx1;
    const float wy1 = iy - y0f, wy0 = 1.0f - wy1;

    const bool inb_x0 = (x0f >= 0.0f) && (x0f <= (float)(Ww - 1));
    const bool inb_x1 = (x1f >= 0.0f) && (x1f <= (float)(Ww - 1));
    const bool inb_y0 = (y0f >= 0.0f) && (y0f <= (float)(Hh - 1));
    const bool inb_y1 = (y1f >= 0.0f) && (y1f <= (float)(Hh - 1));

    const int xi0 = (int)fminf(fmaxf(x0f, 0.0f), (float)(Ww - 1));
    const int xi1 = (int)fminf(fmaxf(x1f, 0.0f), (float)(Ww - 1));
    const int yi0 = (int)fminf(fmaxf(y0f, 0.0f), (float)(Hh - 1));
    const int yi1 = (int)fminf(fmaxf(y1f, 0.0f), (float)(Hh - 1));

    // zeros padding: mask out-of-bounds corner contributions
    const float w00 = wy0 * wx0 * ((inb_y0 && inb_x0) ? 1.0f : 0.0f);
    const float w01 = wy0 * wx1 * ((inb_y0 && inb_x1) ? 1.0f : 0.0f);
    const float w10 = wy1 * wx0 * ((inb_y1 && inb_x0) ? 1.0f : 0.0f);
    const float w11 = wy1 * wx1 * ((inb_y1 && inb_x1) ? 1.0f : 0.0f);

    const int o00 = yi0 * Ww + xi0;
    const int o01 = yi0 * Ww + xi1;
    const int o10 = yi1 * Ww + xi0;
    const int o11 = yi1 * Ww + xi1;

    const float* __restrict__ xb = x + (size_t)n * Cc * HW;
    float* __restrict__ ob = out + (size_t)n * Cc * HW + (size_t)ho * Ww + wo;

#pragma unroll 8
    for (int c = 0; c < Cc; ++c) {
        const float* __restrict__ xc = xb + (size_t)c * HW;
        // one light prefetch per 8-channel unroll group (gfx1250 global_prefetch_b8)
        if (((c & 7) == 0) && (c + 8 < Cc)) {
            __builtin_prefetch(xc + (size_t)8 * HW + o00, 0, 3);
            __builtin_prefetch(xc + (size_t)8 * HW + o10, 0, 3);
        }
        const float v00 = xc[o00];
        const float v01 = xc[o01];
        const float v10 = xc[o10];
        const float v11 = xc[o11];
        float v = v11 * w11;
        v = fmaf(v10, w10, v);
        v = fmaf(v01, w01, v);
        v = fmaf(v00, w00, v);
        // NT store: output is streamed, never re-read; keep x resident in L2
        __builtin_nontemporal_store(v, ob + (size_t)c * HW);
    }
}

extern "C" void kernel_launch(void* const* d_in, const int* in_sizes, int n_in,
                              void* d_out, int out_size, void* d_ws, size_t ws_size,
                              hipStream_t stream) {
    (void)in_sizes; (void)n_in; (void)out_size; (void)d_ws; (void)ws_size;
    const float* x    = (const float*)d_in[0];   // [8,64,256,256]
    const float* grid = (const float*)d_in[1];   // [8,256,256,2]
    float* out        = (float*)d_out;           // [8,64,256,256]

    dim3 blocks(Nn * Hh);   // 2048 blocks: one per (n, ho) output row
    dim3 threads(Ww);       // 256 lanes = 8 wave32 per block
    grid_sample_bilinear_zeros<<<blocks, threads, 0, stream>>>(x, grid, out);
}